// ContextAwareDAC_25460566130982
// MI455X (gfx1250) — compile-verified
//
#include <hip/hip_runtime.h>
#include <math.h>

// ---------------------------------------------------------------------------
// CDNA5 (gfx1250) f32 WMMA: D(16x16) = A(16x4) x B(4x16) + C, full f32.
// A: lane m=lane&15, k = (lane>>4)*2 + comp
// B: lane n=lane&15, k = (lane>>4)*2 + comp
// C/D: lane n=lane&15, VGPR v -> m = v + 8*(lane>>4)
// ---------------------------------------------------------------------------
typedef float v2f __attribute__((ext_vector_type(2)));
typedef float v8f __attribute__((ext_vector_type(8)));

__device__ __forceinline__ v8f wmma_f32_16x16x4(v2f a, v2f b, v8f c) {
  return __builtin_amdgcn_wmma_f32_16x16x4_f32(false, a, false, b, (short)0, c,
                                               false, false);
}

__device__ __forceinline__ float sigmoidf_(float x) {
  return 1.0f / (1.0f + expf(-x));
}

// -------- async global->LDS staging (CDNA5), with portable fallback --------
#if defined(__has_builtin)
#if __has_builtin(__builtin_amdgcn_global_load_async_to_lds_b32) && \
    __has_builtin(__builtin_amdgcn_s_wait_asynccnt)
#define USE_ASYNC_LDS 1
#endif
#endif
#ifndef USE_ASYNC_LDS
#define USE_ASYNC_LDS 0
#endif

__device__ __forceinline__ void stage_f32(const float* g, float* l) {
#if USE_ASYNC_LDS
  __builtin_amdgcn_global_load_async_to_lds_b32(
      (__attribute__((address_space(1))) int*)g,
      (__attribute__((address_space(3))) int*)l, 0, 0);
#else
  *l = *g;
#endif
}

__device__ __forceinline__ void wait_async_stage() {
#if USE_ASYNC_LDS
  __builtin_amdgcn_s_wait_asynccnt(0);
#endif
}

// ---------------------------------------------------------------------------
// Implicit-im2col conv + bias + relu as WMMA GEMM.
// K is padded per input-channel: 25 -> 28 (KPI), ICC=4 channels per chunk
// (Kc = 112). Staging iterates (row, ic, kh, kw) structurally => no integer
// division anywhere; inner loop is pure LDS b64 loads + v_wmma.
// Block: 256 threads = 8 waves; rows blockIdx.x*128..+127; cols 16*blockIdx.y.
// Dynamic LDS: A[128][114] + B[16][114] + rowbase[128] = 66176 B.
// ---------------------------------------------------------------------------
#define ICC 4
#define KPI 28
#define KC 112
#define ASTRIDE 114   // padded row stride: 57 b64-pairs, odd -> conflict-free
#define NCHUNK 22     // ceil(87/4)

__global__ void __launch_bounds__(256)
conv_im2col_wmma(const float* __restrict__ inp, const float* __restrict__ w,
                 const float* __restrict__ bias, float* __restrict__ out,
                 int Hin, int Win, int Hout, int Wout, int M) {
  extern __shared__ float cmem[];
  float* As = cmem;                        // [128][ASTRIDE]
  float* Bs = cmem + 128 * ASTRIDE;        // [16][ASTRIDE]
  int* rowbase = (int*)(cmem + 144 * ASTRIDE);

  const int tid  = threadIdx.x;
  const int lane = tid & 31;
  const int wave = tid >> 5;
  const int l15  = lane & 15;
  const int kb   = (lane >> 4) * 2;
  const int nbase = blockIdx.y * 16;
  const int HinWin = Hin * Win;

  // Per-row base offset into the input image (one divide per row, once).
  if (tid < 128) {
    int m = blockIdx.x * 128 + tid;
    if (m >= M) m = M - 1;
    int ow = m % Wout;
    int oh = (m / Wout) % Hout;
    int ni = m / (Wout * Hout);
    rowbase[tid] = (ni * 87 * Hin + oh) * Win + ow;
  }
  // Static K-padding slots (25..27 of each 28-segment) zeroed once.
  for (int i = tid; i < 128 * ICC * 3; i += 256) {
    int r = i / (ICC * 3), s = i - r * (ICC * 3);
    int icl = s / 3, q = 25 + (s - icl * 3);
    As[r * ASTRIDE + icl * KPI + q] = 0.f;
  }
  for (int i = tid; i < 16 * ICC * 3; i += 256) {
    int n = i / (ICC * 3), s = i - n * (ICC * 3);
    int icl = s / 3, q = 25 + (s - icl * 3);
    Bs[n * ASTRIDE + icl * KPI + q] = 0.f;
  }
  __syncthreads();

  v8f acc = {0.f, 0.f, 0.f, 0.f, 0.f, 0.f, 0.f, 0.f};

  for (int chunk = 0; chunk < NCHUNK; ++chunk) {
    const int icbase = chunk * ICC;
    // ---- stage A: 512 (row,icl) segments, 2 per thread, 25 elems each ----
    for (int p = tid; p < 512; p += 256) {
      int r = p >> 2, icl = p & 3;
      int ic = icbase + icl;
      float* dst = As + r * ASTRIDE + icl * KPI;
      if (ic < 87) {
        const float* src = inp + (size_t)rowbase[r] + (size_t)ic * HinWin;
        for (int kh = 0; kh < 5; ++kh)
          for (int kw = 0; kw < 5; ++kw)
            stage_f32(src + kh * Win + kw, dst + kh * 5 + kw);
      } else {
        for (int q = 0; q < 25; ++q) dst[q] = 0.f;
      }
    }
    // ---- stage B: 64 (col,icl) segments ----
    if (tid < 64) {
      int n = tid >> 2, icl = tid & 3;
      int ic = icbase + icl;
      int oc = nbase + n;
      float* dst = Bs + n * ASTRIDE + icl * KPI;
      if (ic < 87 && oc < 87) {
        const float* src = w + (size_t)oc * 2175 + ic * 25;
        for (int q = 0; q < 25; ++q) stage_f32(src + q, dst + q);
      } else {
        for (int q = 0; q < 25; ++q) dst[q] = 0.f;
      }
    }
    wait_async_stage();
    __syncthreads();
    // ---- compute: 28 wmma per wave, pure LDS b64 + v_wmma ----
    const float* arow = As + (wave * 16 + l15) * ASTRIDE;
    const float* brow = Bs + l15 * ASTRIDE;
    for (int k = 0; k < KC; k += 4) {
      v2f a = *(const v2f*)(arow + k + kb);
      v2f b = *(const v2f*)(brow + k + kb);
      acc = wmma_f32_16x16x4(a, b, acc);
    }
    __syncthreads();
  }

  const int ncol = nbase + l15;
  if (ncol < 87) {
    float bv = bias[ncol];
    for (int v = 0; v < 8; ++v) {
      int m = blockIdx.x * 128 + wave * 16 + v + 8 * (lane >> 4);
      if (m < M) {
        int ow2 = m % Wout;
        int oh2 = (m / Wout) % Hout;
        int ni  = m / (Wout * Hout);
        float y = acc[v] + bv;
        y = (y > 0.f) ? y : 0.f;  // fused relu
        out[(((size_t)ni * 87 + ncol) * Hout + oh2) * Wout + ow2] = y;
      }
    }
  }
}

// 3x3 stride-2 VALID max-pool, (NC,Hin,Win) -> (NC,Hout,Wout)
__global__ void maxpool3x3s2(const float* __restrict__ in, float* __restrict__ out,
                             int Hin, int Win, int Hout, int Wout, int total) {
  int idx = blockIdx.x * blockDim.x + threadIdx.x;
  if (idx >= total) return;
  int ow = idx % Wout;
  int oh = (idx / Wout) % Hout;
  int nc = idx / (Wout * Hout);
  const float* p = in + ((size_t)nc * Hin + oh * 2) * Win + ow * 2;
  float m = -3.4e38f;
  for (int kh = 0; kh < 3; ++kh)
    for (int kw = 0; kw < 3; ++kw)
      m = fmaxf(m, p[kh * Win + kw]);
  out[idx] = m;
}

// pool2 over (256*87, 4, 18) -> 1x8, written into rnn_in[..., 50+ow] (stride 60)
__global__ void pool2_to_rnnin(const float* __restrict__ in,
                               float* __restrict__ rnn_in, int total) {
  int idx = blockIdx.x * blockDim.x + threadIdx.x;
  if (idx >= total) return;  // 256*87*8
  int ow = idx & 7;
  int nc = idx >> 3;
  const float* p = in + (size_t)nc * 4 * 18 + ow * 2;
  float m = -3.4e38f;
  for (int kh = 0; kh < 3; ++kh)
    for (int kw = 0; kw < 3; ++kw)
      m = fmaxf(m, p[kh * 18 + kw]);
  rnn_in[(size_t)nc * 60 + 50 + ow] = m;
  if (ow == 0) {  // zero the K padding slots
    rnn_in[(size_t)nc * 60 + 58] = 0.f;
    rnn_in[(size_t)nc * 60 + 59] = 0.f;
  }
}

// token_embeddings (256,87,50) -> rnn_in[..., 0:50] (stride 60)
__global__ void copy_tokens(const float* __restrict__ tok,
                            float* __restrict__ rnn_in, int total) {
  int idx = blockIdx.x * blockDim.x + threadIdx.x;
  if (idx >= total) return;  // 256*87*50
  int e = idx % 50;
  int nt = idx / 50;
  rnn_in[(size_t)nt * 60 + e] = tok[idx];
}

// ---------------------------------------------------------------------------
// Utterance BiGRU: one persistent 1024-thread workgroup per direction.
// Weights LDS-resident in natural row-major with padded strides (62 / 130:
// odd b64-pair stride => conflict-free): 288 KB of the 320 KB WGP LDS.
// Per step: 768 WMMA tiles (gi: K=58 padded to 60, gh: K=128), then gate
// fusion; __syncthreads is the step barrier.
// ---------------------------------------------------------------------------
#define WIH_S 62
#define WHH_S 130

__global__ void __launch_bounds__(1024)
utt_gru(const float* __restrict__ rnn_in,
        const float* __restrict__ wih_f, const float* __restrict__ whh_f,
        const float* __restrict__ bih_f, const float* __restrict__ bhh_f,
        const float* __restrict__ wih_b, const float* __restrict__ whh_b,
        const float* __restrict__ bih_b, const float* __restrict__ bhh_b,
        float* __restrict__ gi_ws, float* __restrict__ gh_ws,
        float* __restrict__ h_ws, float* __restrict__ Hall) {
  extern __shared__ float smem[];
  const int tid  = threadIdx.x;
  const int lane = tid & 31;
  const int wave = tid >> 5;
  const int dir  = blockIdx.x;

  const float* wih = dir ? wih_b : wih_f;
  const float* whh = dir ? whh_b : whh_f;
  const float* bih = dir ? bih_b : bih_f;
  const float* bhh = dir ? bhh_b : bhh_f;
  float* gi = gi_ws + (size_t)dir * 256 * 384;
  float* gh = gh_ws + (size_t)dir * 256 * 384;
  float* hA = h_ws + (size_t)dir * 2 * 256 * 128;
  float* hB = hA + 256 * 128;

  float* wihL = smem;                  // [384][WIH_S], cols 58..61 zero
  float* whhL = smem + 384 * WIH_S;    // [384][WHH_S], cols 128..129 pad

  for (int i = tid; i < 384 * WIH_S; i += 1024) {
    int j = i / WIH_S, k = i - j * WIH_S;
    wihL[i] = (k < 58) ? wih[(size_t)j * 58 + k] : 0.f;
  }
  for (int i = tid; i < 384 * WHH_S; i += 1024) {
    int j = i / WHH_S, k = i - j * WHH_S;
    whhL[i] = (k < 128) ? whh[(size_t)j * 128 + k] : 0.f;
  }
  for (int i = tid; i < 256 * 128; i += 1024) hA[i] = 0.f;
  __threadfence();
  __syncthreads();

  float* hcur = hA;
  float* hnext = hB;
  const int kb  = (lane >> 4) * 2;
  const int l15 = lane & 15;

  for (int t = 0; t < 87; ++t) {
    const int teff = dir ? (86 - t) : t;
    if (t + 1 < 87) {  // prefetch next step's activations (global_prefetch_b8)
      int tnext = dir ? (85 - t) : (t + 1);
      if (tid < 256)
        __builtin_prefetch(rnn_in + ((size_t)tid * 87 + tnext) * 60, 0, 1);
    }
    // 768 tiles: ids 0..383 -> gi GEMM (K=60 padded), 384..767 -> gh (K=128)
    for (int it = 0; it < 24; ++it) {
      int id = wave + it * 32;            // wave-uniform
      int isH = (id >= 384);
      int tile = isH ? (id - 384) : id;
      int mt = tile / 24, nt = tile - mt * 24;
      int mrow = mt * 16 + l15;
      int ncol = nt * 16 + l15;
      v8f acc = {0.f, 0.f, 0.f, 0.f, 0.f, 0.f, 0.f, 0.f};
      if (!isH) {
        const float* xrow = rnn_in + ((size_t)mrow * 87 + teff) * 60;
        const float* wrow = wihL + (size_t)ncol * WIH_S;
        for (int k = 0; k < 60; k += 4) {
          v2f a = *(const v2f*)(xrow + k + kb);
          v2f b = *(const v2f*)(wrow + k + kb);
          acc = wmma_f32_16x16x4(a, b, acc);
        }
        float bv = bih[ncol];
        for (int v = 0; v < 8; ++v) {
          int m = mt * 16 + v + 8 * (lane >> 4);
          gi[(size_t)m * 384 + ncol] = acc[v] + bv;
        }
      } else {
        const float* xrow = hcur + (size_t)mrow * 128;
        const float* wrow = whhL + (size_t)ncol * WHH_S;
        for (int k = 0; k < 128; k += 4) {
          v2f a = *(const v2f*)(xrow + k + kb);
          v2f b = *(const v2f*)(wrow + k + kb);
          acc = wmma_f32_16x16x4(a, b, acc);
        }
        float bv = bhh[ncol];
        for (int v = 0; v < 8; ++v) {
          int m = mt * 16 + v + 8 * (lane >> 4);
          gh[(size_t)m * 384 + ncol] = acc[v] + bv;
        }
      }
    }
    __threadfence();
    __syncthreads();
    // gate fusion: h' = (1-z)*n + z*h
    for (int i = tid; i < 256 * 128; i += 1024) {
      int m = i >> 7, j = i & 127;
      const float* gim = gi + (size_t)m * 384;
      const float* ghm = gh + (size_t)m * 384;
      float r = sigmoidf_(gim[j] + ghm[j]);
      float z = sigmoidf_(gim[128 + j] + ghm[128 + j]);
      float n = tanhf(gim[256 + j] + r * ghm[256 + j]);
      hnext[i] = (1.f - z) * n + z * hcur[i];
    }
    __threadfence();
    __syncthreads();
    float* tmp = hcur; hcur = hnext; hnext = tmp;
  }
  for (int i = tid; i < 256 * 128; i += 1024) {
    int m = i >> 7, j = i & 127;
    Hall[(size_t)m * 256 + dir * 128 + j] = hcur[i];
  }
}

// ---------------------------------------------------------------------------
// ctx scan: 256 dependent outer steps, each with attention math + a 128-step
// batch-2 bidirectional GRU. 32768 dependent micro-steps => latency bound;
// single persistent 1024-thread workgroup, all state in LDS, VALU dots.
// ---------------------------------------------------------------------------
__global__ void __launch_bounds__(1024)
ctx_scan(const float* __restrict__ Hall,
         const float* __restrict__ ctx_w1, const float* __restrict__ ctx_w2,
         const float* __restrict__ ctx_b2, const float* __restrict__ ctx_w3,
         const float* __restrict__ proj_w, const float* __restrict__ proj_b,
         const float* __restrict__ gx0,
         const float* __restrict__ cg_wih_f, const float* __restrict__ cg_whh_f,
         const float* __restrict__ cg_bih_f, const float* __restrict__ cg_bhh_f,
         const float* __restrict__ cg_wih_b, const float* __restrict__ cg_whh_b,
         const float* __restrict__ cg_bih_b, const float* __restrict__ cg_bhh_b,
         float* __restrict__ feats) {
  __shared__ float Hx[256];
  __shared__ float av[128];
  __shared__ float tb[256];
  __shared__ float Sv[256];
  __shared__ float Av[256];
  __shared__ float oned[256];
  __shared__ float hF[256], hBd[256];
  __shared__ float gxA[256], gxB[256];
  __shared__ float gH[1536];
  __shared__ float red[1024];
  const int tid = threadIdx.x;

  if (tid < 256) {
    gxA[tid] = gx0[tid];        // gx[0] : (2,128)
    gxB[tid] = gx0[256 + tid];  // gx[1]
  }
  __syncthreads();

  const float pb = proj_b[0];

  for (int s = 0; s < 256; ++s) {
    if (tid < 256) Hx[tid] = Hall[(size_t)s * 256 + tid];
    __syncthreads();
    red[tid] = (tid < 256) ? Hx[tid] * proj_w[tid] : 0.f;
    __syncthreads();
    for (int off = 512; off > 0; off >>= 1) {
      if (tid < off) red[tid] += red[tid + off];
      __syncthreads();
    }
    const float hp = red[0];
    __syncthreads();
    if (tid < 128) {
      float acc = 0.f;
      const float* wr = ctx_w1 + (size_t)tid * 256;
      for (int k = 0; k < 256; ++k) acc += Hx[k] * wr[k];
      av[tid] = acc;
    }
    __syncthreads();
    if (tid < 256) {
      int d = tid >> 7, i = tid & 127;
      float acc = ctx_b2[i];
      const float* wr = ctx_w2 + (size_t)i * 128;
      const float* g = gxA + d * 128;
      for (int k = 0; k < 128; ++k) acc += g[k] * wr[k];
      tb[tid] = tanhf(av[i] + acc);
    }
    __syncthreads();
    if (tid < 256) {
      int d = tid >> 7, i = tid & 127;
      float acc = 0.f;
      const float* wr = ctx_w3 + (size_t)i * 128;
      const float* tbd = tb + d * 128;
      for (int k = 0; k < 128; ++k) acc += tbd[k] * wr[k];
      Sv[tid] = acc;
    }
    __syncthreads();
    red[tid] = (tid < 256) ? Sv[tid] : 0.f;
    __syncthreads();
    for (int off = 64; off > 0; off >>= 1) {
      if (tid < 256 && (tid & 127) < off)
        red[tid] = fmaxf(red[tid], red[tid + off]);
      __syncthreads();
    }
    if (tid < 256) Av[tid] = expf(Sv[tid] - red[(tid >> 7) * 128]);
    __syncthreads();
    red[tid] = (tid < 256) ? Av[tid] : 0.f;
    __syncthreads();
    for (int off = 64; off > 0; off >>= 1) {
      if (tid < 256 && (tid & 127) < off) red[tid] += red[tid + off];
      __syncthreads();
    }
    if (tid < 256) {
      float A_ = Av[tid] / red[(tid >> 7) * 128];
      oned[tid] = A_ * hp + pb;   // one_d[d][i] = A[d][i]*(Hx.proj_w) + proj_b
      hF[tid] = gxA[tid];         // fwd init = gx[0]
      hBd[tid] = gxB[tid];        // bwd init = gx[1]
    }
    __syncthreads();
    for (int ttp = 0; ttp < 128; ++ttp) {
      for (int c = tid; c < 1536; c += 1024) {
        int dirx = c / 768;
        int rem = c - dirx * 768;
        int d = rem / 384;
        int j = rem - d * 384;
        const float* hrow = (dirx ? hBd : hF) + d * 128;
        const float* wr = (dirx ? cg_whh_b : cg_whh_f) + (size_t)j * 128;
        float acc = (dirx ? cg_bhh_b : cg_bhh_f)[j];
        for (int k = 0; k < 128; k += 4)
          acc += hrow[k] * wr[k] + hrow[k + 1] * wr[k + 1] +
                 hrow[k + 2] * wr[k + 2] + hrow[k + 3] * wr[k + 3];
        gH[c] = acc;
      }
      __syncthreads();
      if (tid < 512) {
        int dirx = tid >> 8;
        int d = (tid >> 7) & 1;
        int j = tid & 127;
        const float* wihp = dirx ? cg_wih_b : cg_wih_f;
        const float* bihp = dirx ? cg_bih_b : cg_bih_f;
        float x = oned[d * 128 + (dirx ? (127 - ttp) : ttp)];
        const float* gHb = gH + dirx * 768 + d * 384;
        float gr = x * wihp[j] + bihp[j] + gHb[j];
        float gz = x * wihp[128 + j] + bihp[128 + j] + gHb[128 + j];
        float gn = x * wihp[256 + j] + bihp[256 + j];
        float hn = gHb[256 + j];
        float r = sigmoidf_(gr);
        float z = sigmoidf_(gz);
        float n = tanhf(gn + r * hn);
        float* hrow = (dirx ? hBd : hF) + d * 128;
        hrow[j] = (1.f - z) * n + z * hrow[j];
      }
      __syncthreads();
    }
    if (tid < 512) {
      float v = (tid < 256) ? hF[tid] : hBd[tid - 256];
      feats[(size_t)s * 512 + tid] = v;
    }
    if (tid < 256) { gxA[tid] = hF[tid]; gxB[tid] = hBd[tid]; }
    __syncthreads();
  }
}

// ---------------------------------------------------------------------------
// Y = act(X @ W^T + b), W:(Nout,K), K % 4 == 0. One WMMA tile per wave;
// contiguous b64 operand loads.
// ---------------------------------------------------------------------------
__global__ void __launch_bounds__(256)
gemm_xwt(const float* __restrict__ X, const float* __restrict__ W,
         const float* __restrict__ bias, float* __restrict__ Y,
         int M, int K, int Nout, int leaky) {
  const int tid = threadIdx.x;
  const int lane = tid & 31;
  const int wave = tid >> 5;
  const int Ntiles = (Nout + 15) >> 4;
  const int Mtiles = (M + 15) >> 4;
  const int tile = blockIdx.x * 8 + wave;
  if (tile >= Mtiles * Ntiles) return;  // wave-uniform, no barriers below
  const int mt = tile / Ntiles, nt = tile - mt * Ntiles;
  int mrow = mt * 16 + (lane & 15);
  if (mrow >= M) mrow = M - 1;
  const int ncol = nt * 16 + (lane & 15);
  const int ncl = (ncol < Nout) ? ncol : (Nout - 1);
  const int kb = (lane >> 4) * 2;
  const float* xr = X + (size_t)mrow * K + kb;
  const float* wr = W + (size_t)ncl * K + kb;
  v8f acc = {0.f, 0.f, 0.f, 0.f, 0.f, 0.f, 0.f, 0.f};
  for (int k = 0; k < K; k += 4) {
    v2f a = *(const v2f*)(xr + k);
    v2f b = *(const v2f*)(wr + k);
    acc = wmma_f32_16x16x4(a, b, acc);
  }
  if (ncol < Nout) {
    float bv = bias[ncol];
    for (int v = 0; v < 8; ++v) {
      int m = mt * 16 + v + 8 * (lane >> 4);
      if (m < M) {
        float y = acc[v] + bv;
        if (leaky) y = (y > 0.f) ? y : 0.01f * y;
        Y[(size_t)m * Nout + ncol] = y;
      }
    }
  }
}

// ---------------------------------------------------------------------------
extern "C" void kernel_launch(void* const* d_in, const int* in_sizes, int n_in,
                              void* d_out, int out_size, void* d_ws,
                              size_t ws_size, hipStream_t stream) {
  (void)in_sizes; (void)n_in; (void)out_size; (void)ws_size;
  const float* tok    = (const float*)d_in[0];
  const float* tch    = (const float*)d_in[1];
  const float* c1w    = (const float*)d_in[2];
  const float* c1b    = (const float*)d_in[3];
  const float* c2w    = (const float*)d_in[4];
  const float* c2b    = (const float*)d_in[5];
  const float* wih_f  = (const float*)d_in[6];
  const float* whh_f  = (const float*)d_in[7];
  const float* bih_f  = (const float*)d_in[8];
  const float* bhh_f  = (const float*)d_in[9];
  const float* wih_b  = (const float*)d_in[10];
  const float* whh_b  = (const float*)d_in[11];
  const float* bih_b  = (const float*)d_in[12];
  const float* bhh_b  = (const float*)d_in[13];
  const float* ctx_w1 = (const float*)d_in[14];
  const float* ctx_w2 = (const float*)d_in[15];
  const float* ctx_b2 = (const float*)d_in[16];
  const float* ctx_w3 = (const float*)d_in[17];
  const float* proj_w = (const float*)d_in[18];
  const float* proj_b = (const float*)d_in[19];
  const float* gx0    = (const float*)d_in[20];
  const float* cg_wih_f = (const float*)d_in[21];
  const float* cg_whh_f = (const float*)d_in[22];
  const float* cg_bih_f = (const float*)d_in[23];
  const float* cg_bhh_f = (const float*)d_in[24];
  const float* cg_wih_b = (const float*)d_in[25];
  const float* cg_whh_b = (const float*)d_in[26];
  const float* cg_bih_b = (const float*)d_in[27];
  const float* cg_bhh_b = (const float*)d_in[28];
  const float* cls_w1 = (const float*)d_in[29];
  const float* cls_b1 = (const float*)d_in[30];
  const float* cls_w2 = (const float*)d_in[31];
  const float* cls_b2 = (const float*)d_in[32];
  const float* cls_w3 = (const float*)d_in[33];
  const float* cls_b3 = (const float*)d_in[34];

  char* ws = (char*)d_ws;
  size_t off = 0;
  auto alloc = [&](size_t nfl) {
    float* p = (float*)(ws + off);
    off += ((nfl * sizeof(float) + 255) / 256) * 256;
    return p;
  };
  float* conv1out = alloc((size_t)256 * 87 * 17 * 46);  // 69.7 MB
  float* pool1out = alloc((size_t)256 * 87 * 8 * 22);   // 15.7 MB
  float* conv2out = alloc((size_t)256 * 87 * 4 * 18);   //  6.4 MB
  float* rnn_in   = alloc((size_t)256 * 87 * 60);       //  5.3 MB (stride 60)
  float* gi_ws    = alloc((size_t)2 * 256 * 384);
  float* gh_ws    = alloc((size_t)2 * 256 * 384);
  float* h_ws     = alloc((size_t)4 * 256 * 128);
  float* Hall     = alloc((size_t)256 * 256);
  float* feats    = alloc((size_t)256 * 512);
  float* h1       = alloc((size_t)256 * 256);
  float* h2       = alloc((size_t)256 * 128);

  const size_t convLds = (size_t)(144 * ASTRIDE + 128) * sizeof(float); // 66176
  const size_t gruLds  = (size_t)(384 * WIH_S + 384 * WHH_S) * sizeof(float);

  // 1. token embeddings -> rnn_in[:, :, 0:50]
  copy_tokens<<<(256 * 87 * 50 + 255) / 256, 256, 0, stream>>>(
      tok, rnn_in, 256 * 87 * 50);
  // 2. conv1 + relu (WMMA implicit im2col GEMM: 200192 x 2436pad x 87)
  {
    int M = 256 * 17 * 46;  // 200192, multiple of 128
    dim3 g((M + 127) / 128, 6);
    conv_im2col_wmma<<<g, 256, convLds, stream>>>(tch, c1w, c1b, conv1out,
                                                  21, 50, 17, 46, M);
  }
  // 3. pool1
  maxpool3x3s2<<<(256 * 87 * 8 * 22 + 255) / 256, 256, 0, stream>>>(
      conv1out, pool1out, 17, 46, 8, 22, 256 * 87 * 8 * 22);
  // 4. conv2 + relu
  {
    int M = 256 * 4 * 18;   // 18432, multiple of 128
    dim3 g((M + 127) / 128, 6);
    conv_im2col_wmma<<<g, 256, convLds, stream>>>(pool1out, c2w, c2b, conv2out,
                                                  8, 22, 4, 18, M);
  }
  // 5. pool2 -> rnn_in[:, :, 50:58] (+ zero pad 58..59)
  pool2_to_rnnin<<<(256 * 87 * 8 + 255) / 256, 256, 0, stream>>>(
      conv2out, rnn_in, 256 * 87 * 8);
  // 6. utterance BiGRU (persistent workgroup per direction, LDS weights)
  utt_gru<<<2, 1024, gruLds, stream>>>(
      rnn_in, wih_f, whh_f, bih_f, bhh_f, wih_b, whh_b, bih_b, bhh_b,
      gi_ws, gh_ws, h_ws, Hall);
  // 7. ctx scan (latency-bound persistent workgroup)
  ctx_scan<<<1, 1024, 0, stream>>>(
      Hall, ctx_w1, ctx_w2, ctx_b2, ctx_w3, proj_w, proj_b, gx0,
      cg_wih_f, cg_whh_f, cg_bih_f, cg_bhh_f,
      cg_wih_b, cg_whh_b, cg_bih_b, cg_bhh_b, feats);
  // 8. classifier: 256x512x256 -> 256x256x128 -> 256x128x43
  gemm_xwt<<<(16 * 16 + 7) / 8, 256, 0, stream>>>(feats, cls_w1, cls_b1, h1,
                                                  256, 512, 256, 1);
  gemm_xwt<<<(16 * 8 + 7) / 8, 256, 0, stream>>>(h1, cls_w2, cls_b2, h2,
                                                 256, 256, 128, 1);
  gemm_xwt<<<(16 * 3 + 7) / 8, 256, 0, stream>>>(h2, cls_w3, cls_b3,
                                                 (float*)d_out, 256, 128, 43, 0);
}